// GridCell_1760936591659
// MI455X (gfx1250) — compile-verified
//
#include <hip/hip_runtime.h>

// MI455X / gfx1250: circular 9x9 conv x2 + mask select, banded-GEMM on
// V_WMMA_F32_16X16X4_F32 (fp32 matrix pipe), zero-padded banded weight table,
// async global->LDS staging. Each wave computes a 16x32 output strip (two
// 16x16 WMMA tiles) so the shared banded-A operands amortize over 4 WMMAs.

#define Hdim   4096
#define Wdim   4096
#define TILE_H 64
#define TILE_W 128
#define LDS_H  (TILE_H + 8)   // 72
#define LDS_W  (TILE_W + 8)   // 136 (even stride -> conflict-free B reads)
#define NTHREADS 512          // 16 waves: 4 row-tiles x 4 col-strips of 16x32
#define WROW   12             // padded filter row length (16B-aligned rows)
#define WROWS  40             // band rows dy = -15..24 -> 0..39 (zero outside 15..23)
#define WOFF   15

typedef float v2f __attribute__((ext_vector_type(2)));
typedef float v4f __attribute__((ext_vector_type(4)));
typedef float v8f __attribute__((ext_vector_type(8)));

#if __has_builtin(__builtin_amdgcn_global_load_async_to_lds_b32)
#define HAVE_ASYNC_LDS 1
typedef __attribute__((address_space(1))) int* gptr_i32;
typedef __attribute__((address_space(3))) int* lptr_i32;
#else
#define HAVE_ASYNC_LDS 0
#endif

__global__ __launch_bounds__(NTHREADS)
void circ_conv_mask_wmma(const float* __restrict__ x,
                         const float* __restrict__ wdir,
                         const float* __restrict__ wstat,
                         const int*  __restrict__ mask,
                         float* __restrict__ out)
{
    __shared__ __align__(16) float xt[LDS_H * LDS_W];   // wrap-padded input tile
    __shared__ __align__(16) float wz[2][WROWS * WROW]; // zero-padded banded filters

    const int tid  = threadIdx.x;
    const int row0 = blockIdx.y * TILE_H;
    const int col0 = blockIdx.x * TILE_W;

    // ---- Stage 72x136 halo tile with circular wrap addressing (async DMA) ----
    for (int e = tid; e < LDS_H * LDS_W; e += NTHREADS) {
        const int lr = e / LDS_W;
        const int lc = e - lr * LDS_W;
        const int gr = (row0 - 4 + lr) & (Hdim - 1);
        const int gc = (col0 - 4 + lc) & (Wdim - 1);
#if HAVE_ASYNC_LDS
        __builtin_amdgcn_global_load_async_to_lds_b32(
            (gptr_i32)(x + gr * Wdim + gc),
            (lptr_i32)(&xt[e]),
            0, 0);
#else
        xt[e] = x[gr * Wdim + gc];
#endif
    }

    // ---- Build zero-padded banded weight tables ----
    for (int e = tid; e < 2 * WROWS * WROW; e += NTHREADS)
        ((float*)wz)[e] = 0.f;
    __syncthreads();
    if (tid < 162) {
        const int f = (tid >= 81);
        const int t = tid - 81 * f;
        const int r = t / 9;
        const int c = t - 9 * r;
        wz[f][(r + WOFF) * WROW + c] = f ? wstat[t] : wdir[t];
    }
#if HAVE_ASYNC_LDS
#if __has_builtin(__builtin_amdgcn_s_wait_asynccnt)
    __builtin_amdgcn_s_wait_asynccnt(0);
#else
    asm volatile("s_wait_asynccnt 0" ::: "memory");
#endif
#endif
    __syncthreads();

    // ---- Each wave computes one 16x32 output strip (two 16x16 tiles) ----
    const int lane = tid & 31;
    const int wv   = tid >> 5;       // 0..15
    const int wr   = wv >> 2;        // 0..3 (row tile)
    const int wc   = wv & 3;         // 0..3 (32-wide col strip)
    const int h    = lane >> 4;      // lane half (ISA operand striping)
    const int l    = lane & 15;
    const int tr0  = wr * 16;
    const int tc0  = wc * 32;

    v8f acc_d0 = {0.f,0.f,0.f,0.f,0.f,0.f,0.f,0.f};
    v8f acc_s0 = acc_d0;
    v8f acc_d1 = acc_d0;
    v8f acc_s1 = acc_d0;

    // out[r][c] = sum_dx sum_k A_dx[r][k] * xt[k][c+dx],
    // A_dx[r][k] = w[k-r][dx], zero outside the band (padded table handles it).
    #pragma unroll
    for (int q = 0; q < 6; ++q) {
        const int k    = 4 * q + 2 * h;           // K index held by v0 in this half
        const int base = (k - l + WOFF) * WROW;   // banded row for v0 (M = l)

        // Hoist both filters' A values (rows dy0, dy0+1 = 24 floats) to registers.
        const v4f* pd = (const v4f*)&wz[0][base];
        const v4f* ps = (const v4f*)&wz[1][base];
        v4f adv[6], asv[6];
        #pragma unroll
        for (int t = 0; t < 6; ++t) { adv[t] = pd[t]; asv[t] = ps[t]; }

        const float* brow = &xt[(tr0 + k) * LDS_W + tc0 + l];

        #pragma unroll
        for (int dx = 0; dx < 9; ++dx) {
            v2f b0, b1;                    // B: v0 = row k, v1 = row k+1
            b0.x = brow[dx];
            b0.y = brow[LDS_W + dx];
            b1.x = brow[16 + dx];
            b1.y = brow[LDS_W + 16 + dx];

            const int j0 = dx;             // constant after unroll
            const int j1 = dx + WROW;
            v2f ad, as2;
            ad.x  = adv[j0 >> 2][j0 & 3];
            ad.y  = adv[j1 >> 2][j1 & 3];
            as2.x = asv[j0 >> 2][j0 & 3];
            as2.y = asv[j1 >> 2][j1 & 3];

            acc_d0 = __builtin_amdgcn_wmma_f32_16x16x4_f32(
                         false, ad,  false, b0, (short)0, acc_d0, false, false);
            acc_s0 = __builtin_amdgcn_wmma_f32_16x16x4_f32(
                         false, as2, false, b0, (short)0, acc_s0, false, false);
            acc_d1 = __builtin_amdgcn_wmma_f32_16x16x4_f32(
                         false, ad,  false, b1, (short)0, acc_d1, false, false);
            acc_s1 = __builtin_amdgcn_wmma_f32_16x16x4_f32(
                         false, as2, false, b1, (short)0, acc_s1, false, false);
        }
    }

    // ---- D layout: VGPR i -> M = i + 8*h, N = l. Mask-select on store ----
    const int orow  = row0 + tr0 + 8 * h;
    const int ocol0 = col0 + tc0 + l;
    #pragma unroll
    for (int i = 0; i < 8; ++i) {
        const int g0 = (orow + i) * Wdim + ocol0;
        const int g1 = g0 + 16;
        out[g0] = mask[g0] ? acc_d0[i] : acc_s0[i];
        out[g1] = mask[g1] ? acc_d1[i] : acc_s1[i];
    }
}

extern "C" void kernel_launch(void* const* d_in, const int* in_sizes, int n_in,
                              void* d_out, int out_size, void* d_ws, size_t ws_size,
                              hipStream_t stream) {
    const float* x  = (const float*)d_in[0];   // (1,1,4096,4096) fp32
    const float* wd = (const float*)d_in[1];   // (1,1,9,9) fp32
    const float* ws = (const float*)d_in[2];   // (1,1,9,9) fp32
    const int*   m  = (const int*)  d_in[3];   // (4096,4096) bool->int
    float* out = (float*)d_out;                // (1,1,4096,4096) fp32

    dim3 grid(Wdim / TILE_W, Hdim / TILE_H);   // 32 x 64 workgroups
    circ_conv_mask_wmma<<<grid, NTHREADS, 0, stream>>>(x, wd, ws, m, out);
}